// LorentzPrototypeClassifier_87686052315198
// MI455X (gfx1250) — compile-verified
//
#include <hip/hip_runtime.h>
#include <math.h>

typedef __attribute__((ext_vector_type(2))) float v2f;
typedef __attribute__((ext_vector_type(8))) float v8f;

#define DDIM   4096
#define NFINE  100
#define NSUPER 20
#define NTILES 7          // ceil(100/16) -> padded to 112 cols
#define NPAD   (NTILES * 16)
#define MTILE  128        // rows per workgroup (8 waves x 16)
#define KC     32         // K chunk per LDS stage
#define STR    36         // LDS row stride (dwords) for A and B: 16B-aligned, 36*m distinct mod 64

// async 16B copy: global (SGPR base + 32-bit byte offset) -> LDS (byte offset), ASYNCcnt-tracked
__device__ __forceinline__ void async_cp16(unsigned lds_byte_off, unsigned glob_byte_off,
                                           const float* __restrict__ base) {
    asm volatile("global_load_async_to_lds_b128 %0, %1, %2 offset:0"
                 :
                 : "v"(lds_byte_off), "v"(glob_byte_off), "s"(base)
                 : "memory");
}

__device__ __forceinline__ unsigned lds_off(const void* p) {
    return (unsigned)(size_t)p;   // low 32 bits of generic pointer == LDS offset
}

__device__ __forceinline__ void stage_a(float* __restrict__ Al,
                                        const float* __restrict__ cls,
                                        long m0, int k0, int tid) {
    #pragma unroll
    for (int it = 0; it < (MTILE * (KC / 4)) / 256; ++it) {   // 4 iters
        const int idx = tid + it * 256;
        const int m = idx >> 3;       // row in tile
        const int q = idx & 7;        // 16B quad within 32-float chunk
        async_cp16(lds_off(&Al[m * STR + q * 4]),
                   (unsigned)(((m0 + m) * DDIM + k0 + q * 4) * 4), cls);
    }
}

// B stored row-major per column n (same orientation as global!) -> no transpose needed.
__device__ __forceinline__ void stage_b(float* __restrict__ Bl,
                                        const float* __restrict__ proto,
                                        int k0, int tid) {
    for (int idx = tid; idx < NPAD * (KC / 4); idx += 256) {  // 896 copies
        const int n = idx >> 3;
        const int q = idx & 7;
        if (n < NFINE)                // EXEC-masked: pad columns stay zero-filled
            async_cp16(lds_off(&Bl[n * STR + q * 4]),
                       (unsigned)((((long)(NSUPER + n)) * DDIM + k0 + q * 4) * 4), proto);
    }
}

__global__ __launch_bounds__(256)
void lorentz_proto_wmma_kernel(const float* __restrict__ cls,
                               const float* __restrict__ proto,
                               const float* __restrict__ logT,
                               float* __restrict__ out)
{
    __shared__ float Alds[2][MTILE * STR];   // 2 x 18432 B
    __shared__ float Blds[2][NPAD  * STR];   // 2 x 16128 B  (~69 KB of 320 KB WGP LDS)

    const int tid  = threadIdx.x;
    const int wave = tid >> 5;            // 0..7  (wave32)
    const int lane = tid & 31;
    const int half = lane >> 4;           // 0: lanes 0-15, 1: lanes 16-31
    const int nr   = lane & 15;
    const long m0  = (long)blockIdx.x * MTILE;

    v8f acc[NTILES];
    {
        const v8f zero = {0.f, 0.f, 0.f, 0.f, 0.f, 0.f, 0.f, 0.f};
        #pragma unroll
        for (int t = 0; t < NTILES; ++t) acc[t] = zero;
    }

    // prologue: zero-fill B pad columns (never touched again), kick off buffer 0
    for (int idx = tid; idx < 2 * (NPAD - NFINE) * KC; idx += 256) {
        const int b = idx / ((NPAD - NFINE) * KC);
        const int r = idx % ((NPAD - NFINE) * KC);
        Blds[b][(NFINE + r / KC) * STR + (r % KC)] = 0.f;
    }
    stage_a(Alds[0], cls, m0, 0, tid);
    stage_b(Blds[0], proto, 0, tid);
    asm volatile("s_wait_asynccnt 0x0" ::: "memory");
    __syncthreads();

    int buf = 0;
    for (int k0 = 0; k0 < DDIM; k0 += KC) {
        const int nxt = buf ^ 1;
        // issue next chunk's async copies first: they run under the WMMA block
        if (k0 + KC < DDIM) {
            stage_a(Alds[nxt], cls, m0, k0 + KC, tid);
            stage_b(Blds[nxt], proto, k0 + KC, tid);
        }

        const float* Al = Alds[buf];
        const float* Bl = Blds[buf];
        // 8 micro-k x 7 N-tiles = 56 v_wmma_f32_16x16x4_f32
        #pragma unroll
        for (int kk = 0; kk < KC; kk += 4) {
            // A frag (16x4): lane m=nr; VGPR0=K(kk+2*half), VGPR1=K+1 -> ds_load_b64
            const v2f a = *reinterpret_cast<const v2f*>(
                &Al[(wave * 16 + nr) * STR + kk + half * 2]);
            #pragma unroll
            for (int t = 0; t < NTILES; ++t) {
                // B frag (4x16): lane n=t*16+nr; {K,K+1} contiguous -> ds_load_b64
                const v2f b = *reinterpret_cast<const v2f*>(
                    &Bl[(t * 16 + nr) * STR + kk + half * 2]);
                acc[t] = __builtin_amdgcn_wmma_f32_16x16x4_f32(
                    /*neg_a=*/false, a, /*neg_b=*/false, b,
                    /*c_mod=*/(short)0, acc[t],
                    /*reuse_a=*/false, /*reuse_b=*/false);
            }
        }
        asm volatile("s_wait_asynccnt 0x0" ::: "memory");  // nxt staging complete
        __syncthreads();                                   // visible to all waves
        buf = nxt;
    }

    // ---- epilogue: Lorentz correction + distance + temperature ----
    const float T = log1pf(expf(logT[0])) + 1e-3f;   // softplus(log_T) + EPS_T

    float a0[8];
    #pragma unroll
    for (int r = 0; r < 8; ++r) {
        const long i = m0 + wave * 16 + r + half * 8;
        a0[r] = cls[i * DDIM];                        // time component of row i
    }

    #pragma unroll
    for (int t = 0; t < NTILES; ++t) {
        const int  j     = t * 16 + nr;
        const bool valid = (j < NFINE);
        const float b0   = valid ? proto[(long)(NSUPER + j) * DDIM] : 0.f;
        #pragma unroll
        for (int r = 0; r < 8; ++r) {
            const long  i     = m0 + wave * 16 + r + half * 8;
            const float dot   = acc[t][r];
            const float inner = dot - 2.f * a0[r] * b0;   // Lorentz inner product
            const float arg   = -inner;                    // K == 1
            const float u     = fmaxf(arg - 1.f, 0.f);
            const float st    = sqrtf(u * (2.f + u) + 1e-12f);
            const float lg    = log1pf(u + st);
            if (valid)
                out[i * NFINE + j] = -(lg * lg) / T;
        }
    }
}

extern "C" void kernel_launch(void* const* d_in, const int* in_sizes, int n_in,
                              void* d_out, int out_size, void* d_ws, size_t ws_size,
                              hipStream_t stream) {
    const float* cls   = (const float*)d_in[0];   // [16384, 4096] f32
    const float* proto = (const float*)d_in[1];   // [120, 4096]   f32
    const float* logT  = (const float*)d_in[2];   // [1]           f32
    float* out = (float*)d_out;                   // [16384, 100]  f32

    const int M = in_sizes[0] / DDIM;             // 16384
    dim3 grid(M / MTILE);                         // 128 workgroups
    dim3 block(256);                              // 8 waves (wave32)
    lorentz_proto_wmma_kernel<<<grid, block, 0, stream>>>(cls, proto, logT, out);
}